// GeneralConv_57071525430171
// MI455X (gfx1250) — compile-verified
//
#include <hip/hip_runtime.h>
#include <hip/hip_bf16.h>

typedef __attribute__((ext_vector_type(2))) float v2f;
typedef __attribute__((ext_vector_type(8))) float v8f;

#define T_TYPES 4
#define R_REL 4
#define H_HEADS 8
#define DIM 128
#define LSTRIDE 132   // padded LDS row stride (floats) -> 4-bank shift per row
#define WPB 4         // waves (16-node tiles) per block in GEMM kernels

__device__ __forceinline__ void lds_fence() {
  asm volatile("s_wait_dscnt 0" ::: "memory");
}
__device__ __forceinline__ void async_fence() {
  asm volatile("s_wait_asynccnt 0" ::: "memory");
}

// CDNA5 async copy: 16B per lane, global -> LDS, tracked by ASYNCcnt
__device__ __forceinline__ void async_copy_b128(unsigned lds_addr,
                                                unsigned long long gaddr) {
  asm volatile("global_load_async_to_lds_b128 %0, %1, off"
               :
               : "v"(lds_addr), "v"(gaddr)
               : "memory");
}

// order-preserving float<->uint encoding for atomic max over signed floats
__device__ __forceinline__ unsigned fenc(float f) {
  unsigned b = __float_as_uint(f);
  return (b & 0x80000000u) ? ~b : (b | 0x80000000u);
}
__device__ __forceinline__ float fdec(unsigned e) {
  unsigned b = (e & 0x80000000u) ? (e ^ 0x80000000u) : ~e;
  return __uint_as_float(b);
}

__device__ __forceinline__ v8f wmma4(v2f a, v2f b, v8f c) {
  // V_WMMA_F32_16X16X4_F32: (neg_a, A, neg_b, B, c_mod, C, reuse_a, reuse_b)
  return __builtin_amdgcn_wmma_f32_16x16x4_f32(false, a, false, b, (short)0, c,
                                               false, false);
}

// D(16x16) = Xl(16 x 128, LDS row stride LSTRIDE) * W(128x128 row-major)[:, n0:n0+16]
__device__ __forceinline__ v8f gemm_tile(const float* Xl, const float* W,
                                         int n0, int half, int l16) {
  v8f acc = {0.f, 0.f, 0.f, 0.f, 0.f, 0.f, 0.f, 0.f};
#pragma unroll 4
  for (int k0 = 0; k0 < DIM; k0 += 4) {
    int kk = k0 + 2 * half;
    v2f a = *(const v2f*)&Xl[l16 * LSTRIDE + kk];     // A[m][k], m=l16
    v2f b;
    b.x = W[(size_t)kk * DIM + n0 + l16];             // B[k][n], n=l16
    b.y = W[(size_t)(kk + 1) * DIM + n0 + l16];
    acc = wmma4(a, b, acc);
  }
  return acc;
}

// out[n][r][h*16+col] = sum_k Tl[m][h*16+k] * rel[r][h][k][col]   (16x16x16 per (r,h))
__device__ __forceinline__ void rel_apply(const float* Tl, const float* rel,
                                          float* out, const int* nid,
                                          int half, int l16) {
  for (int r = 0; r < R_REL; ++r) {
    for (int h = 0; h < H_HEADS; ++h) {
      const float* M = rel + ((size_t)(r * H_HEADS + h)) * 256;
      v8f acc = {0.f, 0.f, 0.f, 0.f, 0.f, 0.f, 0.f, 0.f};
#pragma unroll
      for (int k0 = 0; k0 < 16; k0 += 4) {
        int kk = k0 + 2 * half;
        v2f a = *(const v2f*)&Tl[l16 * LSTRIDE + h * 16 + kk];
        v2f b;
        b.x = M[kk * 16 + l16];
        b.y = M[(kk + 1) * 16 + l16];
        acc = wmma4(a, b, acc);
      }
#pragma unroll
      for (int j = 0; j < 8; ++j) {
        int m = j + 8 * half;
        int n = nid[m];
        if (n >= 0)
          out[((size_t)n * R_REL + r) * DIM + h * 16 + l16] = acc[j];
      }
    }
  }
}

// ---------------- bucketing ----------------
__global__ void k_init(float* aggr, unsigned* mmax, float* ssum, int* cnt,
                       int N) {
  int idx = blockIdx.x * blockDim.x + threadIdx.x;
  size_t na = (size_t)N * DIM;
  if ((size_t)idx < na) aggr[idx] = 0.f;
  if (idx < N * H_HEADS) { mmax[idx] = 0u; ssum[idx] = 0.f; }
  if (idx < T_TYPES) cnt[idx] = 0;
}

__global__ void k_count(const int* node_type, int* cnt, int N) {
  int n = blockIdx.x * blockDim.x + threadIdx.x;
  if (n < N) atomicAdd(&cnt[node_type[n]], 1);
}

__global__ void k_scan(const int* cnt, int* cursor, int* ends, int* meta,
                       int* tile_type, int* tile_start, int* tile_end) {
  int base = 0, nt = 0;
  for (int t = 0; t < T_TYPES; ++t) {
    int c = cnt[t];
    cursor[t] = base;
    int end = base + c;
    ends[t] = end;
    int tiles = (c + 15) >> 4;
    for (int i = 0; i < tiles; ++i) {
      tile_type[nt] = t;
      tile_start[nt] = base + 16 * i;
      tile_end[nt] = end;
      ++nt;
    }
    base = end;
  }
  meta[0] = nt;
}

__global__ void k_fill(const int* node_type, int* cursor, int* perm, int N) {
  int n = blockIdx.x * blockDim.x + threadIdx.x;
  if (n < N) {
    int pos = atomicAdd(&cursor[node_type[n]], 1);
    perm[pos] = n;
  }
}

// ---------------- projections + per-relation transforms (WMMA) ----------------
__global__ __launch_bounds__(32 * WPB) void k_proj(
    const float* __restrict__ x, const float* __restrict__ Wk,
    const float* __restrict__ bk, const float* __restrict__ Wq,
    const float* __restrict__ bq, const float* __restrict__ Wv,
    const float* __restrict__ bv, const float* __restrict__ rel_att,
    const float* __restrict__ rel_msg, const int* __restrict__ tile_type,
    const int* __restrict__ tile_start, const int* __restrict__ tile_end,
    const int* __restrict__ meta, const int* __restrict__ perm,
    float* __restrict__ Q, float* __restrict__ Krel, float* __restrict__ Vrel) {
  __shared__ float Xs[WPB][16 * LSTRIDE];
  __shared__ float Ts[WPB][16 * LSTRIDE];
  __shared__ int Ns[WPB][16];

  int wave = threadIdx.x >> 5, lane = threadIdx.x & 31;
  int tid = blockIdx.x * WPB + wave;
  if (tid >= meta[0]) return;  // wave-uniform exit: EXEC stays all-ones below

  int t = tile_type[tid], st = tile_start[tid], en = tile_end[tid];
  float* X = Xs[wave];
  float* Tl = Ts[wave];
  int* nid = Ns[wave];
  int half = lane >> 4, l16 = lane & 15;

  const size_t wt = (size_t)t * DIM * DIM;
  // warm L2/WGP$ for the weight B-fragment stream (global_prefetch_b8)
  __builtin_prefetch(Wk + wt + (size_t)threadIdx.x * DIM, 0, 2);
  __builtin_prefetch(Wv + wt + (size_t)threadIdx.x * DIM, 0, 2);
  __builtin_prefetch(Wq + wt + (size_t)threadIdx.x * DIM, 0, 2);

  if (lane < 16) {
    int g = st + lane;
    nid[lane] = (g < en) ? perm[g] : -1;
  }
  lds_fence();

  // stage 16 gathered node rows into LDS with CDNA5 async copies:
  // 32 lanes x B128 = one 512B row per issue, no VGPR round-trip (ASYNCcnt)
  unsigned xbase = (unsigned)(size_t)X + (unsigned)(lane * 16);
  for (int r = 0; r < 16; ++r) {
    int n = nid[r];
    if (n < 0) n = 0;
    unsigned long long ga =
        (unsigned long long)(size_t)(x + (size_t)n * DIM) +
        (unsigned long long)(lane * 16);
    async_copy_b128(xbase + (unsigned)(r * (LSTRIDE * 4)), ga);
  }
  async_fence();
  lds_fence();

  // ---- K = X*Wk[t] + bk[t]  -> staged in LDS -> K_rel via rel_att ----
  for (int n0 = 0; n0 < DIM; n0 += 16) {
    v8f acc = gemm_tile(X, Wk + wt, n0, half, l16);
    float bias = bk[t * DIM + n0 + l16];
#pragma unroll
    for (int j = 0; j < 8; ++j)
      Tl[(j + 8 * half) * LSTRIDE + n0 + l16] = acc[j] + bias;
  }
  lds_fence();
  rel_apply(Tl, rel_att, Krel, nid, half, l16);
  lds_fence();

  // ---- V = X*Wv[t] + bv[t]  -> staged in LDS -> V_rel via rel_msg ----
  for (int n0 = 0; n0 < DIM; n0 += 16) {
    v8f acc = gemm_tile(X, Wv + wt, n0, half, l16);
    float bias = bv[t * DIM + n0 + l16];
#pragma unroll
    for (int j = 0; j < 8; ++j)
      Tl[(j + 8 * half) * LSTRIDE + n0 + l16] = acc[j] + bias;
  }
  lds_fence();
  rel_apply(Tl, rel_msg, Vrel, nid, half, l16);

  // ---- Q = X*Wq[t] + bq[t]  -> straight to global (scattered rows) ----
  for (int n0 = 0; n0 < DIM; n0 += 16) {
    v8f acc = gemm_tile(X, Wq + wt, n0, half, l16);
    float bias = bq[t * DIM + n0 + l16];
#pragma unroll
    for (int j = 0; j < 8; ++j) {
      int m = j + 8 * half;
      int n = nid[m];
      if (n >= 0) Q[(size_t)n * DIM + n0 + l16] = acc[j] + bias;
    }
  }
}

// ---------------- edge passes ----------------
__global__ __launch_bounds__(256) void k_att1(
    const int* __restrict__ ei, const int* __restrict__ et,
    const float* __restrict__ Q, const float* __restrict__ Krel,
    const float* __restrict__ rel_pri, float* __restrict__ att,
    unsigned* __restrict__ mmax, int E) {
  int wave = threadIdx.x >> 5, lane = threadIdx.x & 31;
  int e = blockIdx.x * 8 + wave;
  if (e >= E) return;
  int src = ei[e], tgt = ei[(size_t)E + e], r = et[e];
  float4 q = *(const float4*)&Q[(size_t)tgt * DIM + lane * 4];
  float4 k = *(const float4*)&Krel[((size_t)src * R_REL + r) * DIM + lane * 4];
  float d = q.x * k.x + q.y * k.y + q.z * k.z + q.w * k.w;
  d += __shfl_xor(d, 1, 32);
  d += __shfl_xor(d, 2, 32);  // 4 lanes per head (dk=16)
  int h = lane >> 2;
  if ((lane & 3) == 0) {
    float a = d * rel_pri[r * H_HEADS + h] * 0.25f;  // 1/sqrt(dk)
    att[(size_t)e * H_HEADS + h] = a;
    atomicMax(&mmax[(size_t)tgt * H_HEADS + h], fenc(a));
  }
}

__global__ __launch_bounds__(256) void k_att2(
    const int* __restrict__ ei, float* __restrict__ att,
    const unsigned* __restrict__ mmax, float* __restrict__ ssum, int E) {
  int idx = blockIdx.x * blockDim.x + threadIdx.x;
  if (idx >= E * H_HEADS) return;
  int e = idx >> 3, h = idx & 7;
  int tgt = ei[(size_t)E + e];
  float m = fdec(mmax[(size_t)tgt * H_HEADS + h]);
  float p = __expf(att[idx] - m);
  att[idx] = p;
  atomicAdd(&ssum[(size_t)tgt * H_HEADS + h], p);
}

__global__ __launch_bounds__(256) void k_msg(
    const int* __restrict__ ei, const int* __restrict__ et,
    const float* __restrict__ Vrel, const float* __restrict__ att,
    const float* __restrict__ ssum, float* __restrict__ aggr, int E) {
  int wave = threadIdx.x >> 5, lane = threadIdx.x & 31;
  int e = blockIdx.x * 8 + wave;
  if (e >= E) return;
  int src = ei[e], tgt = ei[(size_t)E + e], r = et[e];
  int h = lane >> 2;
  float a = att[(size_t)e * H_HEADS + h] / ssum[(size_t)tgt * H_HEADS + h];
  float4 v = *(const float4*)&Vrel[((size_t)src * R_REL + r) * DIM + lane * 4];
  float* A = &aggr[(size_t)tgt * DIM + lane * 4];
  atomicAdd(A + 0, v.x * a);
  atomicAdd(A + 1, v.y * a);
  atomicAdd(A + 2, v.z * a);
  atomicAdd(A + 3, v.w * a);
}

// ---------------- update: gelu -> Wa GEMM (WMMA) -> skip -> LayerNorm ----------------
__global__ __launch_bounds__(32 * WPB) void k_update(
    const float* __restrict__ x, const float* __restrict__ aggr,
    const float* __restrict__ Wa, const float* __restrict__ ba,
    const float* __restrict__ skip, const float* __restrict__ gamma,
    const float* __restrict__ beta, const int* __restrict__ tile_type,
    const int* __restrict__ tile_start, const int* __restrict__ tile_end,
    const int* __restrict__ meta, const int* __restrict__ perm,
    float* __restrict__ out) {
  __shared__ float Gs[WPB][16 * LSTRIDE];
  __shared__ float Hs[WPB][16 * LSTRIDE];
  __shared__ int Ns[WPB][16];

  int wave = threadIdx.x >> 5, lane = threadIdx.x & 31;
  int tid = blockIdx.x * WPB + wave;
  if (tid >= meta[0]) return;

  int t = tile_type[tid], st = tile_start[tid], en = tile_end[tid];
  float* G = Gs[wave];
  float* Hl = Hs[wave];
  int* nid = Ns[wave];
  int half = lane >> 4, l16 = lane & 15;

  const float* W = Wa + (size_t)t * DIM * DIM;
  __builtin_prefetch(W + (size_t)threadIdx.x * DIM, 0, 2);

  if (lane < 16) {
    int g = st + lane;
    nid[lane] = (g < en) ? perm[g] : -1;
  }
  lds_fence();

  // gather aggregated messages, apply exact GELU, stage in LDS
  for (int r = 0; r < 16; ++r) {
    int n = nid[r];
    if (n < 0) n = 0;
    float4 v = *(const float4*)&aggr[(size_t)n * DIM + lane * 4];
    v.x = 0.5f * v.x * (1.f + erff(v.x * 0.70710678f));
    v.y = 0.5f * v.y * (1.f + erff(v.y * 0.70710678f));
    v.z = 0.5f * v.z * (1.f + erff(v.z * 0.70710678f));
    v.w = 0.5f * v.w * (1.f + erff(v.w * 0.70710678f));
    *(float4*)&G[r * LSTRIDE + lane * 4] = v;
  }
  lds_fence();

  float alpha = 1.f / (1.f + __expf(-skip[t]));
  for (int n0 = 0; n0 < DIM; n0 += 16) {
    v8f acc = gemm_tile(G, W, n0, half, l16);
    float bias = ba[t * DIM + n0 + l16];
#pragma unroll
    for (int j = 0; j < 8; ++j) {
      int m = j + 8 * half;
      int n = nid[m];
      int ns = (n < 0) ? 0 : n;
      float xv = x[(size_t)ns * DIM + n0 + l16];
      float tr = acc[j] + bias;
      Hl[m * LSTRIDE + n0 + l16] = tr * alpha + xv * (1.f - alpha);
    }
  }
  lds_fence();

  // per-row LayerNorm + per-type affine; lanes 0..15 each own one row
  if (lane < 16) {
    int n = nid[lane];
    if (n >= 0) {
      float s = 0.f, s2 = 0.f;
      for (int c = 0; c < DIM; ++c) {
        float v = Hl[lane * LSTRIDE + c];
        s += v;
        s2 += v * v;
      }
      float mu = s * (1.f / DIM);
      float var = s2 * (1.f / DIM) - mu * mu;
      float rinv = rsqrtf(var + 1e-5f);
      for (int c = 0; c < DIM; ++c) {
        float v = (Hl[lane * LSTRIDE + c] - mu) * rinv;
        out[(size_t)n * DIM + c] = gamma[t * DIM + c] * v + beta[t * DIM + c];
      }
    }
  }
}

// ---------------- host side ----------------
extern "C" void kernel_launch(void* const* d_in, const int* in_sizes, int n_in,
                              void* d_out, int out_size, void* d_ws,
                              size_t ws_size, hipStream_t stream) {
  const float* node_inp = (const float*)d_in[0];
  const int* node_type = (const int*)d_in[1];
  const int* edge_index = (const int*)d_in[2];
  const int* edge_type = (const int*)d_in[3];
  const float* Wk = (const float*)d_in[4];
  const float* bk = (const float*)d_in[5];
  const float* Wq = (const float*)d_in[6];
  const float* bq = (const float*)d_in[7];
  const float* Wv = (const float*)d_in[8];
  const float* bv = (const float*)d_in[9];
  const float* Wa = (const float*)d_in[10];
  const float* ba = (const float*)d_in[11];
  const float* rel_pri = (const float*)d_in[12];
  const float* rel_att = (const float*)d_in[13];
  const float* rel_msg = (const float*)d_in[14];
  const float* skip = (const float*)d_in[15];
  const float* gamma = (const float*)d_in[16];
  const float* beta = (const float*)d_in[17];
  float* out = (float*)d_out;

  const int N = in_sizes[1];
  const int E = in_sizes[3];
  const int MAX_TILES = (N + 15) / 16 + T_TYPES;
  const int MT_PAD = ((MAX_TILES + 3) / 4) * 4;

  // workspace carve-up
  char* w = (char*)d_ws;
  float* Q = (float*)w;      w += (size_t)N * DIM * 4;
  float* Krel = (float*)w;   w += (size_t)N * R_REL * DIM * 4;
  float* Vrel = (float*)w;   w += (size_t)N * R_REL * DIM * 4;
  float* aggr = (float*)w;   w += (size_t)N * DIM * 4;
  float* att = (float*)w;    w += (size_t)E * H_HEADS * 4;
  unsigned* mmax = (unsigned*)w; w += (size_t)N * H_HEADS * 4;
  float* ssum = (float*)w;   w += (size_t)N * H_HEADS * 4;
  int* ib = (int*)w;
  int* cnt = ib + 0;
  int* cursor = ib + 4;
  int* ends = ib + 8;
  int* meta = ib + 16;
  int* tile_type = ib + 32;
  int* tile_start = tile_type + MT_PAD;
  int* tile_end = tile_start + MT_PAD;
  int* perm = tile_end + MT_PAD;

  // 1) init
  {
    size_t total = (size_t)N * DIM;
    int blocks = (int)((total + 255) / 256);
    k_init<<<blocks, 256, 0, stream>>>(aggr, mmax, ssum, cnt, N);
  }
  // 2) histogram node types
  k_count<<<(N + 255) / 256, 256, 0, stream>>>(node_type, cnt, N);
  // 3) scan + tile descriptors
  k_scan<<<1, 1, 0, stream>>>(cnt, cursor, ends, meta, tile_type, tile_start,
                              tile_end);
  // 4) scatter into buckets
  k_fill<<<(N + 255) / 256, 256, 0, stream>>>(node_type, cursor, perm, N);
  // 5) projections + relation transforms (WMMA + async-to-LDS staging)
  k_proj<<<(MAX_TILES + WPB - 1) / WPB, 32 * WPB, 0, stream>>>(
      node_inp, Wk, bk, Wq, bq, Wv, bv, rel_att, rel_msg, tile_type, tile_start,
      tile_end, meta, perm, Q, Krel, Vrel);
  // 6) edge attention logits + segment max
  k_att1<<<(E + 7) / 8, 256, 0, stream>>>(edge_index, edge_type, Q, Krel,
                                          rel_pri, att, mmax, E);
  // 7) exp + segment sum
  k_att2<<<(E * H_HEADS + 255) / 256, 256, 0, stream>>>(edge_index, att, mmax,
                                                        ssum, E);
  // 8) weighted message scatter
  k_msg<<<(E + 7) / 8, 256, 0, stream>>>(edge_index, edge_type, Vrel, att, ssum,
                                         aggr, E);
  // 9) update (WMMA) + LayerNorm
  k_update<<<(MAX_TILES + WPB - 1) / WPB, 32 * WPB, 0, stream>>>(
      node_inp, aggr, Wa, ba, skip, gamma, beta, tile_type, tile_start,
      tile_end, meta, perm, out);
}